// MultiBoxLoss_68719476736651
// MI455X (gfx1250) — compile-verified
//
#include <hip/hip_runtime.h>
#include <hip/hip_bf16.h>
#include <math.h>

// ---------------------------------------------------------------------------
// MultiBoxLoss (SSD300): B=256, N=8732, C=21.  Memory-bound: ~280MB of input
// -> ~12us floor at 23.3 TB/s.  Plan:
//   loc_kernel : 71MB streamed once, masked SmoothL1 + pos count.
//   ce_kernel  : 188MB streamed once by 8732 independent 256-thread blocks
//                (async tile -> LDS -> per-box CE -> 8.9MB ce[] in d_ws).
//   sel_kernel : per batch row, async-copy the CE row into LDS
//                (GLOBAL_LOAD_ASYNC_TO_LDS_B128, ASYNCcnt) and find the
//                sum of the top-k background CEs with a 31-step binary
//                search on float bit patterns (exact under ties).
// Cross-lane reductions use V_WMMA_F32_16X16X4_F32 with an all-ones B matrix
// as a single-instruction wave32 sum.
// A fused fallback (round-1 design, 16B of ws) is kept for small ws_size.
// ---------------------------------------------------------------------------

#define BATCH 256
#define NBOX  8732
#define NCLS  21
#define BN    (BATCH * NBOX)

typedef __attribute__((ext_vector_type(2))) float v2f;
typedef __attribute__((ext_vector_type(8))) float v8f;

#if defined(__AMDGCN__) && __has_builtin(__builtin_amdgcn_global_load_async_to_lds_b128)
#define HAVE_ASYNC_LDS 1
typedef int v4i_vs __attribute__((vector_size(16)));            // matches builtin param
typedef __attribute__((address_space(1))) v4i_vs* as1_v4i;
typedef __attribute__((address_space(3))) v4i_vs* as3_v4i;
__device__ __forceinline__ void async_copy_b128(const void* g, void* l) {
    __builtin_amdgcn_global_load_async_to_lds_b128(
        (as1_v4i)(v4i_vs*)const_cast<void*>(g), (as3_v4i)(v4i_vs*)l, 0, 0);
}
__device__ __forceinline__ void async_wait_all() {
    asm volatile("s_wait_asynccnt 0x0" ::: "memory");
}
#endif

// Full wave32 sum via one WMMA: A(16x4) holds {v,0} per lane, B = ones.
// Sum of the 8 D VGPRs gives the half-wave sums; shfl_xor(16) completes it.
// Requires EXEC == all ones (call only from uniform control flow).
__device__ __forceinline__ float wave_reduce_sum(float v) {
    v2f a; a[0] = v;    a[1] = 0.0f;
    v2f b; b[0] = 1.0f; b[1] = 1.0f;
    v8f c = {};
    v8f d = __builtin_amdgcn_wmma_f32_16x16x4_f32(
        false, a, false, b, (short)0, c, false, false);
    float s = d[0] + d[1] + d[2] + d[3] + d[4] + d[5] + d[6] + d[7];
    s += __shfl_xor(s, 16, 32);
    return s;
}

__device__ __forceinline__ float block_reduce_bcast(float v, float* scratch8, int tid) {
    float s = wave_reduce_sum(v);
    if ((tid & 31) == 0) scratch8[tid >> 5] = s;
    __syncthreads();
    float tot = 0.0f;
#pragma unroll
    for (int w = 0; w < 8; ++w) tot += scratch8[w];
    __syncthreads();
    return tot;
}

__device__ __forceinline__ float smooth_l1(float d) {
    float a = fabsf(d);
    return (a < 1.0f) ? 0.5f * d * d : (a - 0.5f);
}

// ws layout (floats): ws[0]=loc_loss ws[1]=conf_loss ws[2]=num_matched
// ws[16..] = ce[BN] (two-phase path only; 64B offset keeps 16B alignment)

__global__ void init_ws(float* ws) {
    if (threadIdx.x < 4) ws[threadIdx.x] = 0.0f;
}

// --------------------------- loc + num_matched -----------------------------
__global__ void loc_kernel(const float* __restrict__ lp,
                           const float* __restrict__ lt,
                           const int*   __restrict__ tgt,
                           float* __restrict__ ws) {
    __shared__ float scratch[8];
    const int tid    = threadIdx.x;
    const int gid    = blockIdx.x * blockDim.x + tid;
    const int stride = gridDim.x * blockDim.x;
    const float4* lp4 = (const float4*)lp;
    const float4* lt4 = (const float4*)lt;

    float acc = 0.0f, cnt = 0.0f;
    for (int i = gid; i < BN; i += stride) {
        const int   t = tgt[i];
        const float m = (t > 0) ? 1.0f : 0.0f;
        const float4 p = lp4[i];
        const float4 q = lt4[i];
        float s = smooth_l1(p.x - q.x) + smooth_l1(p.y - q.y) +
                  smooth_l1(p.z - q.z) + smooth_l1(p.w - q.w);
        acc += s * m;
        cnt += m;
    }
    const float bacc = block_reduce_bcast(acc, scratch, tid);
    const float bcnt = block_reduce_bcast(cnt, scratch, tid);
    if (tid == 0) { atomicAdd(&ws[0], bacc); atomicAdd(&ws[2], bcnt); }
}

// ------------------- phase 1: per-box cross entropy ------------------------
// One block per 256-box tile; BN/256 = 8732 blocks exactly (no tail).
__global__ void __launch_bounds__(256)
ce_kernel(const float* __restrict__ conf,
          const int*   __restrict__ tgt,
          float* __restrict__ ce_out) {
    __shared__ float stage[256 * NCLS];                 // 21504 B
    const int tid  = threadIdx.x;
    const int base = blockIdx.x * 256;                  // first box of tile

    const float4* src = (const float4*)(conf + (size_t)base * NCLS);
    float4* dst = (float4*)stage;
    for (int j = tid; j < 256 * NCLS / 4; j += 256) {   // 1344 float4
#if defined(HAVE_ASYNC_LDS)
        async_copy_b128(&src[j], &dst[j]);
#else
        dst[j] = src[j];
#endif
    }
#if defined(HAVE_ASYNC_LDS)
    async_wait_all();
#endif
    __syncthreads();

    const float* x = stage + tid * NCLS;                // stride 21: no bank conflicts
    float m = x[0];
#pragma unroll
    for (int j = 1; j < NCLS; ++j) m = fmaxf(m, x[j]);
    float s = 0.0f;
#pragma unroll
    for (int j = 0; j < NCLS; ++j) s += __expf(x[j] - m);
    const int t = tgt[base + tid];
    ce_out[base + tid] = __logf(s) + m - x[t];
}

// ------------- phase 2: per-row hard-negative mining + sums ----------------
__global__ void __launch_bounds__(256)
sel_kernel(const float* __restrict__ ce,
           const int*   __restrict__ tgt,
           float* __restrict__ ws) {
    __shared__ float cl[NBOX];                          // 34928 B
    __shared__ float scratch[8];
    const int b   = blockIdx.x;
    const int tid = threadIdx.x;
    const float* cerow = ce  + (size_t)b * NBOX;        // 16B-aligned (NBOX*4 % 16 == 0)
    const int*   trow  = tgt + (size_t)b * NBOX;

    // Pull the whole CE row into LDS (2183 b128 ops, exact).
    const float4* src4 = (const float4*)cerow;
    float4* cl4 = (float4*)cl;
    for (int j = tid; j < NBOX / 4; j += 256) {
#if defined(HAVE_ASYNC_LDS)
        async_copy_b128(&src4[j], &cl4[j]);
#else
        cl4[j] = src4[j];
#endif
    }
#if defined(HAVE_ASYNC_LDS)
    async_wait_all();
#endif
    __syncthreads();

    // Zero positives in cl, accumulate positive CE + count.
    float pos_ce = 0.0f, pos_cnt = 0.0f;
    for (int i = tid; i < NBOX; i += 256) {
        const int t = trow[i];
        if (t > 0) { pos_ce += cl[i]; pos_cnt += 1.0f; cl[i] = 0.0f; }
    }
    __syncthreads();

    const float npf  = block_reduce_bcast(pos_cnt, scratch, tid);
    const float pces = block_reduce_bcast(pos_ce,  scratch, tid);
    const int   np   = (int)(npf + 0.5f);
    const int   k    = min(3 * np, NBOX - 1);

    float contrib = pces;
    if (k > 0) {                                        // uniform branch
        unsigned tbits = 0u;                            // k-th largest, bitwise
        for (int bit = 30; bit >= 0; --bit) {
            const unsigned trial = tbits | (1u << bit);
            float c = 0.0f;
            for (int i = tid; i < NBOX; i += 256)
                c += (__float_as_uint(cl[i]) >= trial) ? 1.0f : 0.0f;
            const float tot = block_reduce_bcast(c, scratch, tid);
            if ((int)(tot + 0.5f) >= k) tbits = trial;
        }
        const float tval = __uint_as_float(tbits);
        float sgt = 0.0f, cgt = 0.0f;
        for (int i = tid; i < NBOX; i += 256) {
            const float v = cl[i];
            if (v > tval) { sgt += v; cgt += 1.0f; }
        }
        const float tsgt = block_reduce_bcast(sgt, scratch, tid);
        const float tcgt = block_reduce_bcast(cgt, scratch, tid);
        contrib += tsgt + ((float)k - tcgt) * tval;     // exact under ties
    }
    if (tid == 0) atomicAdd(&ws[1], contrib);
}

// -------------------- fused fallback (16B of ws only) ----------------------
__global__ void __launch_bounds__(256)
conf_fused_kernel(const float* __restrict__ conf,
                  const int*   __restrict__ tgt,
                  float* __restrict__ ws) {
    __shared__ float cl[NBOX];
    __shared__ float stage[256 * NCLS];
    __shared__ float scratch[8];
    const int b   = blockIdx.x;
    const int tid = threadIdx.x;
    const float* row  = conf + (size_t)b * NBOX * NCLS;
    const int*   trow = tgt  + (size_t)b * NBOX;

    float pos_ce = 0.0f, pos_cnt = 0.0f;
    for (int base = 0; base < NBOX; base += 256) {
        const int nb  = min(256, NBOX - base);
        const int nf4 = nb * NCLS / 4;
        const float4* src = (const float4*)(row + (size_t)base * NCLS);
        float4* dst = (float4*)stage;
        for (int j = tid; j < nf4; j += 256) dst[j] = src[j];
        __syncthreads();
        if (tid < nb) {
            const float* x = stage + tid * NCLS;
            float m = x[0];
#pragma unroll
            for (int j = 1; j < NCLS; ++j) m = fmaxf(m, x[j]);
            float s = 0.0f;
#pragma unroll
            for (int j = 0; j < NCLS; ++j) s += __expf(x[j] - m);
            const int   t  = trow[base + tid];
            const float v  = __logf(s) + m - x[t];
            const bool  p  = (t > 0);
            cl[base + tid] = p ? 0.0f : v;
            if (p) { pos_ce += v; pos_cnt += 1.0f; }
        }
        __syncthreads();
    }

    const float npf  = block_reduce_bcast(pos_cnt, scratch, tid);
    const float pces = block_reduce_bcast(pos_ce,  scratch, tid);
    const int   np   = (int)(npf + 0.5f);
    const int   k    = min(3 * np, NBOX - 1);

    float contrib = pces;
    if (k > 0) {
        unsigned tbits = 0u;
        for (int bit = 30; bit >= 0; --bit) {
            const unsigned trial = tbits | (1u << bit);
            float c = 0.0f;
            for (int i = tid; i < NBOX; i += 256)
                c += (__float_as_uint(cl[i]) >= trial) ? 1.0f : 0.0f;
            const float tot = block_reduce_bcast(c, scratch, tid);
            if ((int)(tot + 0.5f) >= k) tbits = trial;
        }
        const float tval = __uint_as_float(tbits);
        float sgt = 0.0f, cgt = 0.0f;
        for (int i = tid; i < NBOX; i += 256) {
            const float v = cl[i];
            if (v > tval) { sgt += v; cgt += 1.0f; }
        }
        const float tsgt = block_reduce_bcast(sgt, scratch, tid);
        const float tcgt = block_reduce_bcast(cgt, scratch, tid);
        contrib += tsgt + ((float)k - tcgt) * tval;
    }
    if (tid == 0) atomicAdd(&ws[1], contrib);
}

__global__ void finalize(const float* __restrict__ ws, float* __restrict__ out) {
    const float nm    = ws[2];
    const float denom = fmaxf(nm, 1.0f);
    const float loss  = (ws[0] + ws[1]) / denom;
    out[0] = (nm == 0.0f) ? 0.0f : loss;
}

extern "C" void kernel_launch(void* const* d_in, const int* in_sizes, int n_in,
                              void* d_out, int out_size, void* d_ws, size_t ws_size,
                              hipStream_t stream) {
    const float* loc_preds    = (const float*)d_in[0];
    const float* loc_targets  = (const float*)d_in[1];
    const float* conf_preds   = (const float*)d_in[2];
    const int*   conf_targets = (const int*)d_in[3];
    float* out = (float*)d_out;
    float* ws  = (float*)d_ws;
    (void)in_sizes; (void)n_in; (void)out_size;

    init_ws<<<1, 32, 0, stream>>>(ws);
    loc_kernel<<<2048, 256, 0, stream>>>(loc_preds, loc_targets, conf_targets, ws);

    const size_t need = 64 + (size_t)BN * sizeof(float);
    if (ws_size >= need) {
        float* ce = ws + 16;
        ce_kernel<<<BN / 256, 256, 0, stream>>>(conf_preds, conf_targets, ce);
        sel_kernel<<<BATCH, 256, 0, stream>>>(ce, conf_targets, ws);
    } else {
        conf_fused_kernel<<<BATCH, 256, 0, stream>>>(conf_preds, conf_targets, ws);
    }
    finalize<<<1, 1, 0, stream>>>(ws, out);
}